// Winograd_72988674228739
// MI455X (gfx1250) — compile-verified
//
#include <hip/hip_runtime.h>

// Problem constants (from reference): N=16, C=64, K=64, H=W=130, T=32, P=16384
#define TPD   32            // tiles per spatial dim
#define CCH   64
#define KCH   64
#define HIN   130
#define WOUT  128
#define VPAD  66            // padded C stride in LDS (bank-conflict avoidance)
#define SMEM_FLOATS (36 * 16 * VPAD)

typedef __attribute__((ext_vector_type(2))) float v2f;
typedef __attribute__((ext_vector_type(8))) float v8f;

// Columns of A_T (4x6): ATC[col][row]
__device__ const float ATC[6][4] = {
    {1.f, 0.f, 0.f, 0.f}, {1.f, 1.f, 1.f, 1.f}, {1.f, -1.f, 1.f, -1.f},
    {1.f, 2.f, 4.f, 8.f}, {1.f, -2.f, 4.f, -8.f}, {0.f, 0.f, 0.f, 1.f}};

// y = B_T * x (one 6-vector)
__device__ __forceinline__ void bt6(const float x[6], float y[6]) {
    y[0] =  4.f * x[0] - 5.f * x[2] + x[4];
    y[1] = -4.f * x[1] - 4.f * x[2] + x[3] + x[4];
    y[2] =  4.f * x[1] - 4.f * x[2] - x[3] + x[4];
    y[3] = -2.f * x[1] -       x[2] + 2.f * x[3] + x[4];
    y[4] =  2.f * x[1] -       x[2] - 2.f * x[3] + x[4];
    y[5] =  4.f * x[1] - 5.f * x[3] + x[5];
}

// o = G * [a,b,c]^T  (G is 6x3)
__device__ __forceinline__ void g3to6(float a, float b, float c, float o[6]) {
    o[0] = 0.25f * a;
    o[1] = -(1.f / 6.f) * (a + b + c);
    o[2] = (1.f / 6.f) * (-a + b - c);
    o[3] = (1.f / 24.f) * a + (1.f / 12.f) * b + (1.f / 6.f) * c;
    o[4] = (1.f / 24.f) * a - (1.f / 12.f) * b + (1.f / 6.f) * c;
    o[5] = c;
}

// Filter transform: Ut[e][k][c] = (G w[k][c] G^T)[e],  e = a*6+b.  4096 threads.
__global__ void wino_filter_k(const float* __restrict__ w, float* __restrict__ Ut) {
    int tid = blockIdx.x * blockDim.x + threadIdx.x;
    int k = tid >> 6, c = tid & 63;
    const float* wp = w + (k * CCH + c) * 9;
    float t[6][3];
#pragma unroll
    for (int j = 0; j < 3; ++j) {
        float col[6];
        g3to6(wp[j], wp[3 + j], wp[6 + j], col);
#pragma unroll
        for (int a = 0; a < 6; ++a) t[a][j] = col[a];
    }
#pragma unroll
    for (int a = 0; a < 6; ++a) {
        float u6[6];
        g3to6(t[a][0], t[a][1], t[a][2], u6);
#pragma unroll
        for (int b = 0; b < 6; ++b)
            Ut[(((a * 6 + b) * KCH + k) << 6) + c] = u6[b];
    }
}

// Fused: input transform -> 36x WMMA GEMM (C-reduce) -> output transform.
// 1024 workgroups x 128 threads. WG = 16 consecutive tiles (same n, ty).
__global__ __launch_bounds__(128) void wino_main_k(const float* __restrict__ x,
                                                   const float* __restrict__ Ut,
                                                   float* __restrict__ y) {
    extern __shared__ float smem[];  // V[36][16][VPAD]
    const int wg  = blockIdx.x;
    const int n   = wg >> 6;
    const int rem = wg & 63;
    const int ty  = rem >> 1;
    const int tx0 = (rem & 1) << 4;
    const int tid = threadIdx.x;

    // ---- Phase 1: input transform, 8 (tile,channel) patches per thread ----
#pragma unroll
    for (int it = 0; it < 8; ++it) {
        int pair = tid + (it << 7);
        int t = pair >> 6, c = pair & 63;
        const float* bp = x + ((size_t)(n * CCH + c) * HIN + ty * 4) * HIN + (tx0 + t) * 4;
        float X[6][6];
#pragma unroll
        for (int rr = 0; rr < 6; ++rr) {
            const float* rp = bp + rr * HIN;  // 8B-aligned
            v2f q0 = *(const v2f*)(rp);
            v2f q1 = *(const v2f*)(rp + 2);
            v2f q2 = *(const v2f*)(rp + 4);
            X[rr][0] = q0.x; X[rr][1] = q0.y;
            X[rr][2] = q1.x; X[rr][3] = q1.y;
            X[rr][4] = q2.x; X[rr][5] = q2.y;
        }
        float tmp[6][6];
#pragma unroll
        for (int j = 0; j < 6; ++j) {
            float col[6] = {X[0][j], X[1][j], X[2][j], X[3][j], X[4][j], X[5][j]};
            float o[6];
            bt6(col, o);
#pragma unroll
            for (int i = 0; i < 6; ++i) tmp[i][j] = o[i];
        }
#pragma unroll
        for (int i = 0; i < 6; ++i) {
            float o[6];
            bt6(tmp[i], o);
#pragma unroll
            for (int j = 0; j < 6; ++j)
                smem[((i * 6 + j) * 16 + t) * VPAD + c] = o[j];
        }
    }
    __syncthreads();

    // ---- Phase 2: per-wave 16-tile x 16-k block; loop over 36 tile points ----
    const int lane = tid & 31;
    const int wave = tid >> 5;
    const int kb   = wave << 4;
    const int row  = lane & 15;       // M index (tile) for A; N index (k) for B/D
    const int half = lane >> 4;
    const int off2 = half << 1;       // K-chunk offset within the 4-wide reduce step
    const int kcol = kb + row;

    float Yr[8][4][4] = {};

    for (int i = 0; i < 6; ++i) {
        for (int j = 0; j < 6; ++j) {
            const int e = i * 6 + j;
            v8f acc = {0.f, 0.f, 0.f, 0.f, 0.f, 0.f, 0.f, 0.f};
            const float* ap = smem + (e * 16 + row) * VPAD + off2;           // LDS A
            const float* bp = Ut + (((e * KCH + kcol) << 6)) + off2;          // L2  B
            // warm next tile-point's U block (global_prefetch_b8)
            const int en = (e + 1 < 36) ? e + 1 : 35;
            __builtin_prefetch(Ut + ((en * KCH + kcol) << 6), 0, 0);
#pragma unroll
            for (int kk = 0; kk < 16; ++kk) {  // C reduction: 16 x (16x16x4)
                v2f a = *(const v2f*)(ap + kk * 4);
                v2f b = *(const v2f*)(bp + kk * 4);
                acc = __builtin_amdgcn_wmma_f32_16x16x4_f32(
                    false, a, false, b, (short)0, acc, false, false);
            }
            // fold M_e into output accumulators: Y += A_T[:,i] m A_T[:,j]^T
            float ci[4], cj[4];
#pragma unroll
            for (int q = 0; q < 4; ++q) { ci[q] = ATC[i][q]; cj[q] = ATC[j][q]; }
#pragma unroll
            for (int v = 0; v < 8; ++v) {
                float m = acc[v];
#pragma unroll
                for (int a2 = 0; a2 < 4; ++a2) {
                    float tv = ci[a2] * m;
#pragma unroll
                    for (int b2 = 0; b2 < 4; ++b2)
                        Yr[v][a2][b2] = __builtin_fmaf(tv, cj[b2], Yr[v][a2][b2]);
                }
            }
        }
    }

    // ---- Phase 3: store 4x4 output tiles (aligned float4) ----
#pragma unroll
    for (int v = 0; v < 8; ++v) {
        int t = v + (half << 3);  // D-matrix M index
        float* op = y + ((size_t)(n * KCH + kcol) * WOUT + ty * 4) * WOUT + (tx0 + t) * 4;
#pragma unroll
        for (int a2 = 0; a2 < 4; ++a2) {
            float4 o = make_float4(Yr[v][a2][0], Yr[v][a2][1], Yr[v][a2][2], Yr[v][a2][3]);
            *(float4*)(op + a2 * WOUT) = o;
        }
    }
}

extern "C" void kernel_launch(void* const* d_in, const int* in_sizes, int n_in,
                              void* d_out, int out_size, void* d_ws, size_t ws_size,
                              hipStream_t stream) {
    (void)in_sizes; (void)n_in; (void)out_size; (void)ws_size;
    const float* x = (const float*)d_in[0];
    const float* w = (const float*)d_in[1];
    float* out = (float*)d_out;
    float* Ut  = (float*)d_ws;  // 36*64*64 floats = 576 KB

    wino_filter_k<<<16, 256, 0, stream>>>(w, Ut);

    const size_t shmem = SMEM_FLOATS * sizeof(float);  // ~148.5 KB (<160 KB CU-mode cap)
    (void)hipFuncSetAttribute(reinterpret_cast<const void*>(wino_main_k),
                              hipFuncAttributeMaxDynamicSharedMemorySize, (int)shmem);
    wino_main_k<<<16 * TPD * TPD / 16, 128, shmem, stream>>>(x, Ut, out);
}